// AddSLoss_85289460564354
// MI455X (gfx1250) — compile-verified
//
#include <hip/hip_runtime.h>
#include <hip/hip_bf16.h>
#include <math.h>

typedef __attribute__((ext_vector_type(2))) float v2f;
typedef __attribute__((ext_vector_type(8))) float v8f;

constexpr int BS    = 16;
constexpr int NPTS  = 2048;
constexpr int TILES = NPTS / 16;   // 128 column tiles

// IEEE float -> order-preserving unsigned key (total order, -0 < +0).
__device__ __forceinline__ unsigned int ordkey(float f) {
    unsigned int b = __float_as_uint(f);
    return b ^ (unsigned int)(((int)b >> 31) | 0x80000000);
}

// ---------------------------------------------------------------------------
// Prep: tf = R*mp + t per point; augmented target (x,y,z,|t|^2); zero output.
// tf4.w = -0.5 so the WMMA A-fragment is uniformly (-2 * component):
//   (-2x, -2y, -2z, 1)  ->  score = -2*tf.t + |t|^2  (argmin-equivalent to d2)
// ---------------------------------------------------------------------------
__global__ __launch_bounds__(256) void adds_prep(
    const float* __restrict__ target, const float* __restrict__ mp,
    const float* __restrict__ H,
    float4* __restrict__ tf4, float4* __restrict__ ta4,
    float* __restrict__ out)
{
    int tid = blockIdx.x * blockDim.x + threadIdx.x;
    if (tid < BS) out[tid] = 0.0f;            // harness poisons d_out; re-zero
    if (tid >= BS * NPTS) return;

    int b = tid / NPTS;
    const float* Hb = H + b * 16;             // H[b] row-major 4x4
    float x = mp[tid*3+0], y = mp[tid*3+1], z = mp[tid*3+2];
    // tf[e] = sum_d mp[d] * H[e][d] + H[e][3]   (== mp @ R^T + t)
    float tx = fmaf(x, Hb[0], fmaf(y, Hb[1], fmaf(z, Hb[2],  Hb[3])));
    float ty = fmaf(x, Hb[4], fmaf(y, Hb[5], fmaf(z, Hb[6],  Hb[7])));
    float tz = fmaf(x, Hb[8], fmaf(y, Hb[9], fmaf(z, Hb[10], Hb[11])));
    tf4[tid] = make_float4(tx, ty, tz, -0.5f);

    float gx = target[tid*3+0], gy = target[tid*3+1], gz = target[tid*3+2];
    ta4[tid] = make_float4(gx, gy, gz, fmaf(gx, gx, fmaf(gy, gy, gz*gz)));
}

// ---------------------------------------------------------------------------
// Main: per wave, 16 query rows. Loop 64 iterations x 2 tiles:
//   two independent v_wmma_f32_16x16x4_f32 chains (co-exec with argmin VALU).
// Branchless 64-bit-key butterfly argmin, gather, L2, wave-sum, atomicAdd.
// ---------------------------------------------------------------------------
__global__ __launch_bounds__(256) void adds_main(
    const float* __restrict__ target, const int* __restrict__ idxv,
    const float2* __restrict__ tf2,   // tf4 viewed as float2 pairs
    const float2* __restrict__ ta2,   // ta4 viewed as float2 pairs
    const float4* __restrict__ tf4,
    float* __restrict__ out)
{
    __shared__ float2 sB[NPTS * 2];   // 32 KB: augmented targets for this batch

    const int b    = blockIdx.x >> 4;
    const int qblk = blockIdx.x & 15;
    const int lane = threadIdx.x & 31;
    const int wave = threadIdx.x >> 5;

    // Cooperative stage of batch-b augmented targets into LDS (8x b128/thread)
    {
        const float4* src = (const float4*)(ta2 + (size_t)b * NPTS * 2);
        float4* dst = (float4*)sB;
        for (int i = threadIdx.x; i < NPTS; i += 256) dst[i] = src[i];
    }
    __syncthreads();

    const int col = lane & 15;        // N (and M) position within tile
    const int hi  = lane >> 4;        // lane half: K={0,1} vs K={2,3}
    const int q0  = qblk * 128 + wave * 16;

    // A fragment: lanes 0-15 hold K0,K1; lanes 16-31 hold K2,K3 (M = col)
    float2 aq = tf2[((size_t)b * NPTS + q0 + col) * 2 + hi];
    v2f afrag;
    afrag.x = -2.0f * aq.x;
    afrag.y = -2.0f * aq.y;           // hi half: -2 * (-0.5) = 1.0

    // Two independent argmin chains (even tiles / odd tiles).
    float minv0[8], minv1[8];
    int   mini0[8], mini1[8];
#pragma unroll
    for (int v = 0; v < 8; ++v) {
        minv0[v] = INFINITY; mini0[v] = 0;
        minv1[v] = INFINITY; mini1[v] = 0;
    }

    const v8f czero = {0.f, 0.f, 0.f, 0.f, 0.f, 0.f, 0.f, 0.f};
    for (int t = 0; t < TILES; t += 2) {
        float2 tb0 = sB[((t    ) * 16 + col) * 2 + hi];
        float2 tb1 = sB[((t + 1) * 16 + col) * 2 + hi];
        v2f bf0; bf0.x = tb0.x; bf0.y = tb0.y;
        v2f bf1; bf1.x = tb1.x; bf1.y = tb1.y;

        v8f d0 = __builtin_amdgcn_wmma_f32_16x16x4_f32(
            false, afrag, false, bf0, (short)0, czero, false, false);
        v8f d1 = __builtin_amdgcn_wmma_f32_16x16x4_f32(
            false, afrag, false, bf1, (short)0, czero, false, false);

        const int n0 = (t    ) * 16 + col;
        const int n1 = (t + 1) * 16 + col;
#pragma unroll
        for (int v = 0; v < 8; ++v) {
            if (d0[v] < minv0[v]) { minv0[v] = d0[v]; mini0[v] = n0; }
            if (d1[v] < minv1[v]) { minv1[v] = d1[v]; mini1[v] = n1; }
        }
    }

    // Merge chains + butterfly across the 16 lanes of each half.
    // Key = (ordered-float << 32) | index: unsigned min == argmin with
    // first-occurrence tie-break (matches jnp.argmin).
    unsigned long long kred[8];
#pragma unroll
    for (int v = 0; v < 8; ++v) {
        unsigned long long k0 =
            ((unsigned long long)ordkey(minv0[v]) << 32) | (unsigned int)mini0[v];
        unsigned long long k1 =
            ((unsigned long long)ordkey(minv1[v]) << 32) | (unsigned int)mini1[v];
        unsigned long long kv = (k1 < k0) ? k1 : k0;
#pragma unroll
        for (int off = 8; off >= 1; off >>= 1) {
            unsigned long long ok = __shfl_xor(kv, off);
            kv = (ok < kv) ? ok : kv;
        }
        kred[v] = kv;   // all 16 lanes of the half now agree; row M = v + 8*hi
    }

    // Lanes col<8 each own one query row: M = col + 8*hi, NN idx = kred[col].
    float dist = 0.0f;
    if (col < 8) {
        unsigned int nn = 0;
#pragma unroll
        for (int v = 0; v < 8; ++v)
            nn = (col == v) ? (unsigned int)kred[v] : nn;

        const int m = col + 8 * hi;
        const int q = q0 + m;
        const bool sym = ((unsigned)idxv[b] <= 3u);     // idx in {0,1,2,3}
        const int tn = sym ? (int)nn : q;

        const float* T = target + (size_t)b * NPTS * 3;
        float4 tq = tf4[(size_t)b * NPTS + q];
        float dx = tq.x - T[tn*3+0];
        float dy = tq.y - T[tn*3+1];
        float dz = tq.z - T[tn*3+2];
        dist = sqrtf(fmaf(dx, dx, fmaf(dy, dy, dz*dz)));
    }

    // Wave-wide sum of the 16 active distances, one atomic per wave.
#pragma unroll
    for (int off = 16; off >= 1; off >>= 1) dist += __shfl_xor(dist, off);
    if (lane == 0) atomicAdd(&out[b], dist * (1.0f / NPTS));
}

// ---------------------------------------------------------------------------
extern "C" void kernel_launch(void* const* d_in, const int* in_sizes, int n_in,
                              void* d_out, int out_size, void* d_ws, size_t ws_size,
                              hipStream_t stream)
{
    const float* target = (const float*)d_in[0];   // [16,2048,3] f32
    const float* mp     = (const float*)d_in[1];   // [16,2048,3] f32
    const int*   idxv   = (const int*)  d_in[2];   // [16,1] i32
    const float* H      = (const float*)d_in[3];   // [16,4,4] f32
    float*       out    = (float*)d_out;           // [16] f32

    float4* tf4 = (float4*)d_ws;                   // 16*2048 float4 = 512 KB
    float4* ta4 = tf4 + (size_t)BS * NPTS;         // 512 KB

    adds_prep<<<dim3((BS * NPTS + 255) / 256), dim3(256), 0, stream>>>(
        target, mp, H, tf4, ta4, out);

    adds_main<<<dim3(BS * 16), dim3(256), 0, stream>>>(
        target, idxv, (const float2*)tf4, (const float2*)ta4, tf4, out);
}